// InteractionBlock_2516850835964
// MI455X (gfx1250) — compile-verified
//
#include <hip/hip_runtime.h>
#include <cstddef>

typedef float v2f __attribute__((ext_vector_type(2)));
typedef float v8f __attribute__((ext_vector_type(8)));

#define NN 50000
#define NE 800000
// MUL=32, NATTR=16, NBASIS=8, HID=8

__device__ __forceinline__ v8f wmma4(v2f a, v2f b, v8f c) {
  // D = A(16x4 f32) * B(4x16 f32) + C(16x16 f32)
  return __builtin_amdgcn_wmma_f32_16x16x4_f32(false, a, false, b, (short)0, c,
                                               false, false);
}

// ---------------------------------------------------------------------------
// Repack W (K x 32, row-major) into pair-interleaved layout so the WMMA
// B-operand {W[k][c], W[k+1][c]} is one aligned b64 load:
//   p[(k>>1)*64 + c*2 + (k&1)] = w[k*32 + c]
// ---------------------------------------------------------------------------
__global__ __launch_bounds__(256) void repack_pairs(const float* __restrict__ w,
                                                    float* __restrict__ p,
                                                    int K) {
  const int i = blockIdx.x * 256 + threadIdx.x;  // over K*32
  if (i >= K * 32) return;
  const int k = i >> 5, c = i & 31;
  p[(k >> 1) * 64 + c * 2 + (k & 1)] = w[k * 32 + c];
}

// ---------------------------------------------------------------------------
// Kernel A: one wave per 16-node tile.
//   phase 1: h_s = h_s0 @ W1_s * inv_m ; h_v = h_v0 @ W1_v * inv_m  (to ws)
//   phase 2: sc self-connection, K=512, A built on the fly          (to out)
// ---------------------------------------------------------------------------
__global__ __launch_bounds__(256) void node_prep_kernel(
    const float* __restrict__ x, const float* __restrict__ h,
    const float* __restrict__ W1sp, const float* __restrict__ W1vp,
    const float* __restrict__ Wscsp, const float* __restrict__ Wscvp,
    float* __restrict__ hs_ws, float* __restrict__ hv_ws,
    float* __restrict__ out) {
  const int lane = threadIdx.x & 31;
  const int wave = threadIdx.x >> 5;
  const int tile = blockIdx.x * 8 + wave;
  if (tile >= NN / 16) return;               // wave-uniform guard: EXEC full
  const int node0 = tile * 16;
  const int l15 = lane & 15;
  const int half = lane >> 4;
  const int koff = half * 2;
  const int arow = node0 + l15;
  const float* hrow = h + (size_t)arow * 128;

  float xr[16];
#pragma unroll
  for (int v = 0; v < 16; v += 4) {
    const float4 t = *(const float4*)(x + (size_t)arow * 16 + v);
    xr[v + 0] = t.x; xr[v + 1] = t.y; xr[v + 2] = t.z; xr[v + 3] = t.w;
  }

  // -------- phase 1: h @ W1 (K = 32) --------
  v8f cs0 = {}, cs1 = {};
  v8f cv0[3] = {}, cv1[3] = {};
#pragma unroll
  for (int k0 = 0; k0 < 32; k0 += 4) {
    const int kg = k0 + koff;
    const v2f a_s = *(const v2f*)(hrow + kg);
    const float* bs = W1sp + (kg >> 1) * 64;
    const float* bv = W1vp + (kg >> 1) * 64;
    const v2f b_s0 = *(const v2f*)(bs + l15 * 2);
    const v2f b_s1 = *(const v2f*)(bs + 32 + l15 * 2);
    const v2f b_v0 = *(const v2f*)(bv + l15 * 2);
    const v2f b_v1 = *(const v2f*)(bv + 32 + l15 * 2);
    cs0 = wmma4(a_s, b_s0, cs0);
    cs1 = wmma4(a_s, b_s1, cs1);
#pragma unroll
    for (int m = 0; m < 3; ++m) {
      v2f a_v;
      a_v.x = hrow[32 + kg * 3 + m];
      a_v.y = hrow[32 + (kg + 1) * 3 + m];
      cv0[m] = wmma4(a_v, b_v0, cv0[m]);
      cv1[m] = wmma4(a_v, b_v1, cv1[m]);
    }
  }
  const float inv_m = 0.17677669529663687f;  // 1/sqrt(32)
#pragma unroll
  for (int r = 0; r < 8; ++r) {
    const int node = node0 + r + half * 8;
    hs_ws[node * 32 + l15]      = cs0[r] * inv_m;
    hs_ws[node * 32 + 16 + l15] = cs1[r] * inv_m;
#pragma unroll
    for (int m = 0; m < 3; ++m) {
      hv_ws[node * 96 + l15 * 3 + m]        = cv0[m][r] * inv_m;
      hv_ws[node * 96 + (16 + l15) * 3 + m] = cv1[m][r] * inv_m;
    }
  }

  // -------- phase 2: self-connection (K = 512) --------
  v8f ds0 = {}, ds1 = {};
  v8f dv0[3] = {}, dv1[3] = {};
  for (int u = 0; u < 32; ++u) {
    const float hs_u = hrow[u];
    float hv_u[3];
#pragma unroll
    for (int m = 0; m < 3; ++m) hv_u[m] = hrow[32 + u * 3 + m];
    if (u + 1 < 32) {  // warm L0 for next u's B rows (global_prefetch_b8)
      __builtin_prefetch(Wscsp + (((u + 1) * 16) >> 1) * 64, 0, 3);
      __builtin_prefetch(Wscvp + (((u + 1) * 16) >> 1) * 64, 0, 3);
    }
#pragma unroll
    for (int kk = 0; kk < 4; ++kk) {
      const int v0 = kk * 4 + koff;          // attr index for a.x
      const int kg = u * 16 + v0;            // global K for B operand
      const float* bs = Wscsp + (kg >> 1) * 64;
      const float* bv = Wscvp + (kg >> 1) * 64;
      const v2f b_s0 = *(const v2f*)(bs + l15 * 2);
      const v2f b_s1 = *(const v2f*)(bs + 32 + l15 * 2);
      const v2f b_v0 = *(const v2f*)(bv + l15 * 2);
      const v2f b_v1 = *(const v2f*)(bv + 32 + l15 * 2);
      v2f a_s; a_s.x = hs_u * xr[v0]; a_s.y = hs_u * xr[v0 + 1];
      ds0 = wmma4(a_s, b_s0, ds0);
      ds1 = wmma4(a_s, b_s1, ds1);
#pragma unroll
      for (int m = 0; m < 3; ++m) {
        v2f a_v; a_v.x = hv_u[m] * xr[v0]; a_v.y = hv_u[m] * xr[v0 + 1];
        dv0[m] = wmma4(a_v, b_v0, dv0[m]);
        dv1[m] = wmma4(a_v, b_v1, dv1[m]);
      }
    }
  }
  const float inv_sc = 0.044194173824159216f;  // 1/sqrt(512)
#pragma unroll
  for (int r = 0; r < 8; ++r) {
    const int node = node0 + r + half * 8;
    float* orow = out + (size_t)node * 128;
    orow[l15]      = ds0[r] * inv_sc;
    orow[16 + l15] = ds1[r] * inv_sc;
#pragma unroll
    for (int m = 0; m < 3; ++m) {
      orow[32 + l15 * 3 + m]        = dv0[m][r] * inv_sc;
      orow[32 + (16 + l15) * 3 + m] = dv1[m][r] * inv_sc;
    }
  }
}

// ---------------------------------------------------------------------------
// Kernel B: one wave per 16-EDGE tile. Builds message rows on the fly as the
// WMMA A-operand (lane&15 = edge row, lane>>4 = K sub-offset), applies W2
// BEFORE the scatter (segsum and the linear map commute), then scatters only
// 128 (not 256) f32 atomics per edge into the L2-resident agg buffers.
//   mid channels 0..31  : sA = hs*sh0*wA      -> W2 rows 0..31
//   mid channels 32..63 : sD = (hv.sh1)/√3*wD -> W2 rows 32..63
//   (same for vB / vC against W2_v, per m-plane)
// ---------------------------------------------------------------------------
__global__ __launch_bounds__(256) void edge_kernel(
    const float* __restrict__ ele, const float* __restrict__ esh,
    const int* __restrict__ eidx,
    const float* __restrict__ fc_w1, const float* __restrict__ fc_w2,
    const float* __restrict__ hs_ws, const float* __restrict__ hv_ws,
    const float* __restrict__ W2sp, const float* __restrict__ W2vp,
    float* __restrict__ agg_s, float* __restrict__ agg_v) {
  const int lane = threadIdx.x & 31;
  const int wave = threadIdx.x >> 5;
  const int tile = blockIdx.x * 8 + wave;
  if (tile >= NE / 16) return;               // wave-uniform guard
  const int e0 = tile * 16;
  const int l15 = lane & 15;
  const int half = lane >> 4;
  const int koff = half * 2;
  const int eA = e0 + l15;                   // edge whose A rows this lane owns

  // ---- per-edge scalars for this lane's edge ----
  const int srcA = eidx[NE + eA];
  const float sh0 = esh[(size_t)eA * 4];
  float sh1[3];
#pragma unroll
  for (int m = 0; m < 3; ++m) sh1[m] = esh[(size_t)eA * 4 + 1 + m];

  // ---- edge MLP: z = silu(ele @ fc_w1 / sqrt(8)) / sqrt(8)  (1/sqrt(HID)
  //      folded into z so all w-entries come out pre-scaled) ----
  const float invb = 0.35355339059327373f;   // 1/sqrt(8)
  float el[8];
#pragma unroll
  for (int j = 0; j < 8; ++j) el[j] = ele[(size_t)eA * 8 + j];
  float z[8];
#pragma unroll
  for (int i = 0; i < 8; ++i) {
    float acc = 0.f;
#pragma unroll
    for (int j = 0; j < 8; ++j) acc += el[j] * fc_w1[j * 8 + i];
    acc *= invb;
    z[i] = (acc / (1.f + __expf(-acc))) * invb;  // silu * 1/sqrt(HID)
  }
  // ---- this lane's 16 channels (c = 4j+koff, 4j+koff+1), 4 weight types ----
  float wA[16], wB[16], wC[16], wD[16];
#pragma unroll
  for (int j = 0; j < 8; ++j) {
    const int c = 4 * j + koff;
    v2f a = {0.f, 0.f}, b = {0.f, 0.f}, cc = {0.f, 0.f}, d = {0.f, 0.f};
#pragma unroll
    for (int i = 0; i < 8; ++i) {
      a  += z[i] * *(const v2f*)(fc_w2 + i * 128 + c);
      b  += z[i] * *(const v2f*)(fc_w2 + i * 128 + 32 + c);
      cc += z[i] * *(const v2f*)(fc_w2 + i * 128 + 64 + c);
      d  += z[i] * *(const v2f*)(fc_w2 + i * 128 + 96 + c);
    }
    wA[2 * j] = a.x;  wA[2 * j + 1] = a.y;
    wB[2 * j] = b.x;  wB[2 * j + 1] = b.y;
    wC[2 * j] = cc.x; wC[2 * j + 1] = cc.y;
    wD[2 * j] = d.x;  wD[2 * j + 1] = d.y;
  }

  const float* hsrow = hs_ws + (size_t)srcA * 32;
  const float* hvrow = hv_ws + (size_t)srcA * 96;
  const float inv3 = 0.5773502691896258f;    // 1/sqrt(3)

  v8f cs0 = {}, cs1 = {};
  v8f cv0[3] = {}, cv1[3] = {};
  // ---- K region 1: mid channels 0..31 (sA rows / vB rows) ----
#pragma unroll
  for (int k0 = 0; k0 < 32; k0 += 4) {
    const int kg = k0 + koff;
    const int j2 = (k0 >> 2) * 2;
    const v2f hs = *(const v2f*)(hsrow + kg);
    const float* bs = W2sp + (kg >> 1) * 64;
    const float* bv = W2vp + (kg >> 1) * 64;
    const v2f b_s0 = *(const v2f*)(bs + l15 * 2);
    const v2f b_s1 = *(const v2f*)(bs + 32 + l15 * 2);
    const v2f b_v0 = *(const v2f*)(bv + l15 * 2);
    const v2f b_v1 = *(const v2f*)(bv + 32 + l15 * 2);
    v2f a_s; a_s.x = hs.x * sh0 * wA[j2]; a_s.y = hs.y * sh0 * wA[j2 + 1];
    cs0 = wmma4(a_s, b_s0, cs0);
    cs1 = wmma4(a_s, b_s1, cs1);
#pragma unroll
    for (int m = 0; m < 3; ++m) {
      v2f a_v;
      a_v.x = hs.x * sh1[m] * wB[j2];
      a_v.y = hs.y * sh1[m] * wB[j2 + 1];
      cv0[m] = wmma4(a_v, b_v0, cv0[m]);
      cv1[m] = wmma4(a_v, b_v1, cv1[m]);
    }
  }
  // ---- K region 2: mid channels 32..63 (sD rows / vC rows) ----
#pragma unroll
  for (int k0 = 32; k0 < 64; k0 += 4) {
    const int kg = k0 + koff;
    const int c0 = kg - 32;
    const int j2 = ((k0 - 32) >> 2) * 2;
    float hv0[3], hv1[3];
#pragma unroll
    for (int m = 0; m < 3; ++m) {
      hv0[m] = hvrow[c0 * 3 + m];
      hv1[m] = hvrow[(c0 + 1) * 3 + m];
    }
    const float d0 = (hv0[0] * sh1[0] + hv0[1] * sh1[1] + hv0[2] * sh1[2]) * inv3;
    const float d1 = (hv1[0] * sh1[0] + hv1[1] * sh1[1] + hv1[2] * sh1[2]) * inv3;
    const float* bs = W2sp + (kg >> 1) * 64;
    const float* bv = W2vp + (kg >> 1) * 64;
    const v2f b_s0 = *(const v2f*)(bs + l15 * 2);
    const v2f b_s1 = *(const v2f*)(bs + 32 + l15 * 2);
    const v2f b_v0 = *(const v2f*)(bv + l15 * 2);
    const v2f b_v1 = *(const v2f*)(bv + 32 + l15 * 2);
    v2f a_s; a_s.x = d0 * wD[j2]; a_s.y = d1 * wD[j2 + 1];
    cs0 = wmma4(a_s, b_s0, cs0);
    cs1 = wmma4(a_s, b_s1, cs1);
#pragma unroll
    for (int m = 0; m < 3; ++m) {
      v2f a_v;
      a_v.x = hv0[m] * sh0 * wC[j2];
      a_v.y = hv1[m] * sh0 * wC[j2 + 1];
      cv0[m] = wmma4(a_v, b_v0, cv0[m]);
      cv1[m] = wmma4(a_v, b_v1, cv1[m]);
    }
  }
  // ---- scatter: 128 atomics/edge (W2 already applied) ----
#pragma unroll
  for (int r = 0; r < 8; ++r) {
    const int e = e0 + r + half * 8;         // C/D row r holds this edge
    const int dst = eidx[e];
    atomicAdd(&agg_s[(size_t)dst * 32 + l15], cs0[r]);
    atomicAdd(&agg_s[(size_t)dst * 32 + 16 + l15], cs1[r]);
#pragma unroll
    for (int m = 0; m < 3; ++m) {
      float* plane = agg_v + (size_t)m * NN * 32;
      atomicAdd(&plane[(size_t)dst * 32 + l15], cv0[m][r]);
      atomicAdd(&plane[(size_t)dst * 32 + 16 + l15], cv1[m][r]);
    }
  }
}

// ---------------------------------------------------------------------------
// Kernel C: elementwise finish  out += agg * (1/sqrt(64) * 1/sqrt(16))
// ---------------------------------------------------------------------------
__global__ __launch_bounds__(256) void finish_kernel(
    const float* __restrict__ agg_s, const float* __restrict__ agg_v,
    float* __restrict__ out) {
  const int i = blockIdx.x * 256 + threadIdx.x;  // over NN*32
  if (i >= NN * 32) return;
  const int n = i >> 5, w = i & 31;
  const float scl = 0.03125f;
  float* orow = out + (size_t)n * 128;
  orow[w] += agg_s[i] * scl;
#pragma unroll
  for (int m = 0; m < 3; ++m)
    orow[32 + w * 3 + m] += agg_v[(size_t)m * NN * 32 + i] * scl;
}

extern "C" void kernel_launch(void* const* d_in, const int* in_sizes, int n_in,
                              void* d_out, int out_size, void* d_ws,
                              size_t ws_size, hipStream_t stream) {
  const float* x     = (const float*)d_in[0];
  const float* h     = (const float*)d_in[1];
  const float* ele   = (const float*)d_in[2];
  const float* esh   = (const float*)d_in[3];
  const int*   eidx  = (const int*)d_in[4];
  const float* W1_s  = (const float*)d_in[5];
  const float* W1_v  = (const float*)d_in[6];
  const float* fc_w1 = (const float*)d_in[7];
  const float* fc_w2 = (const float*)d_in[8];
  const float* Wsc_s = (const float*)d_in[9];
  const float* Wsc_v = (const float*)d_in[10];
  const float* W2_s  = (const float*)d_in[11];
  const float* W2_v  = (const float*)d_in[12];
  float* out = (float*)d_out;

  float* ws    = (float*)d_ws;
  float* hs_ws = ws;                                // N*32
  float* hv_ws = hs_ws + (size_t)NN * 32;           // N*96
  float* agg_s = hv_ws + (size_t)NN * 96;           // N*32
  float* agg_v = agg_s + (size_t)NN * 32;           // 3 * N*32 (m-planes)
  float* W1sp  = agg_v + (size_t)NN * 96;           // 32*32
  float* W1vp  = W1sp + 1024;
  float* Wscsp = W1vp + 1024;                       // 512*32
  float* Wscvp = Wscsp + 16384;
  float* W2sp  = Wscvp + 16384;                     // 64*32
  float* W2vp  = W2sp + 2048;

  hipMemsetAsync(agg_s, 0, (size_t)NN * 128 * sizeof(float), stream);

  repack_pairs<<<(32 * 32 + 255) / 256, 256, 0, stream>>>(W1_s, W1sp, 32);
  repack_pairs<<<(32 * 32 + 255) / 256, 256, 0, stream>>>(W1_v, W1vp, 32);
  repack_pairs<<<(512 * 32 + 255) / 256, 256, 0, stream>>>(Wsc_s, Wscsp, 512);
  repack_pairs<<<(512 * 32 + 255) / 256, 256, 0, stream>>>(Wsc_v, Wscvp, 512);
  repack_pairs<<<(64 * 32 + 255) / 256, 256, 0, stream>>>(W2_s, W2sp, 64);
  repack_pairs<<<(64 * 32 + 255) / 256, 256, 0, stream>>>(W2_v, W2vp, 64);

  const int nblkA = (NN / 16 + 7) / 8;              // 3125 tiles, 8 waves/blk
  node_prep_kernel<<<nblkA, 256, 0, stream>>>(x, h, W1sp, W1vp, Wscsp, Wscvp,
                                              hs_ws, hv_ws, out);
  const int nblkB = (NE / 16 + 7) / 8;              // 50000 tiles
  edge_kernel<<<nblkB, 256, 0, stream>>>(ele, esh, eidx, fc_w1, fc_w2, hs_ws,
                                         hv_ws, W2sp, W2vp, agg_s, agg_v);
  finish_kernel<<<(NN * 32 + 255) / 256, 256, 0, stream>>>(agg_s, agg_v, out);
}